// NequipLayer_2259152798307
// MI455X (gfx1250) — compile-verified
//
#include <hip/hip_runtime.h>
#include <hip/hip_bf16.h>
#include <cstdint>

typedef float v2f __attribute__((ext_vector_type(2)));
typedef float v8f __attribute__((ext_vector_type(8)));

#define N_NODES 20000
#define N_EDGES 320000
#define EW 4  // waves per block in edge kernel

__device__ __forceinline__ v8f wmma4(v2f a, v2f b, v8f c) {
  // D(16x16,f32) = A(16x4,f32) * B(4x16,f32) + C
  return __builtin_amdgcn_wmma_f32_16x16x4_f32(false, a, false, b, (short)0, c, false, false);
}
__device__ __forceinline__ float swishf(float x) { return x / (1.0f + __expf(-x)); }
__device__ __forceinline__ v8f vzero() { v8f z = {0.f,0.f,0.f,0.f,0.f,0.f,0.f,0.f}; return z; }

// ---------------------------------------------------------------------------
// K0: repack weights into K-pair layout: pW[(k/2)*C + c] = float2(W[k][c], W[k+1][c])
// so the WMMA B operand (rows k, k+1 at column c) is one contiguous b64 load.
// ---------------------------------------------------------------------------
struct PrepArgs {
  const float* src[9];
  float* dst[9];
  int K[9];
  int C[9];
};

__global__ __launch_bounds__(256) void k_prep(PrepArgs pa) {
  int tid = blockIdx.x * blockDim.x + threadIdx.x;
  int stride = gridDim.x * blockDim.x;
  #pragma unroll 1
  for (int mi = 0; mi < 9; mi++) {
    const float* s = pa.src[mi];
    float* d = pa.dst[mi];
    int C = pa.C[mi];
    int total = (pa.K[mi] >> 1) * C;
    for (int p = tid; p < total; p += stride) {
      int k2 = p / C, c = p - k2 * C;
      d[2*p]     = s[(2*k2)   * C + c];
      d[2*p + 1] = s[(2*k2+1) * C + c];
    }
  }
}

// ---------------------------------------------------------------------------
// K1: per-node linears.  y[:,0:64]=x0@Wl0/8 ; y[:,64+m*64+u]=x1_m@Wl1/8
//     res[:,0:128]      = sum_s diag(attr_s) x0   @ Wr_s[:,s,:] / sqrt(320)
//     res[:,128+m*64+u] = sum_s diag(attr_s) x1_m @ Wr_v[:,s,:] / sqrt(320)
// grid (1250 tiles, 9) x 128 threads -> 36 wave-jobs per node tile.
// A tile staged unit-stride in LDS; B from pair-packed weights (b64 loads).
// ---------------------------------------------------------------------------
__global__ __launch_bounds__(128) void k_node_pre(
    const float* __restrict__ nf, const float* __restrict__ attrs,
    const float* __restrict__ pWl0, const float* __restrict__ pWl1,
    const float* __restrict__ pWrs, const float* __restrict__ pWrv,
    float* __restrict__ y, float* __restrict__ res)
{
  __shared__ float ldsA[4][16][64];
  const int lane = threadIdx.x & 31;
  const int wave = threadIdx.x >> 5;
  const int lr = lane & 15;
  const int half = lane >> 4;
  const int tile = blockIdx.x;
  const int job = blockIdx.y * 4 + wave;   // 0..35
  const int row0 = tile * 16;

  // decode job
  int j, m = 0, aoff, astep, bC, outoff;
  bool isRes = (job >= 16);
  const v2f* Bp;
  if (!isRes) {
    if (job < 4) { j = job; Bp = (const v2f*)pWl0; aoff = 0;      astep = 1; bC = 64; outoff = j*16; }
    else { int t = job - 4; m = t >> 2; j = t & 3;
           Bp = (const v2f*)pWl1; aoff = 64 + m; astep = 3; bC = 64; outoff = 64 + m*64 + j*16; }
  } else {
    int t = job - 16;
    if (t < 8) { j = t; Bp = (const v2f*)pWrs; aoff = 0;      astep = 1; bC = 640; outoff = j*16; }
    else { t -= 8; m = t >> 2; j = t & 3;
           Bp = (const v2f*)pWrv; aoff = 64 + m; astep = 3; bC = 320; outoff = 128 + m*64 + j*16; }
  }

  // stage this wave's A tile (16 rows x 64 K) unit-stride into LDS
  {
    const float* base = nf + row0 * 256 + aoff;
    for (int t = lane; t < 1024; t += 32) {
      int r = t >> 6, u = t & 63;
      ldsA[wave][r][u] = base[r * 256 + u * astep];
    }
  }
  __syncthreads();

  const v2f* arow2 = (const v2f*)&ldsA[wave][lr][0];
  const int col = j * 16 + lr;
  v8f c = vzero();

  if (!isRes) {
    #pragma unroll 4
    for (int t = 0; t < 16; t++) {
      int k2 = half + 2 * t;
      c = wmma4(arow2[k2], Bp[k2 * bC + col], c);
    }
    float* out = y + row0 * 256 + outoff;
    #pragma unroll
    for (int i = 0; i < 8; i++) {
      int orow = i + 8 * half;
      out[orow * 256 + lr] = c[i] * 0.125f;      // 1/sqrt(64)
    }
  } else {
    const float* atr = attrs + (row0 + lr) * 5;
    const int sOff = (bC == 640) ? 128 : 64;
    for (int s = 0; s < 5; s++) {
      float as = atr[s];                          // per-row (per-lane) A scaling
      int colS = s * sOff + col;
      #pragma unroll 4
      for (int t = 0; t < 16; t++) {
        int k2 = half + 2 * t;
        v2f a = arow2[k2];
        a.x *= as; a.y *= as;
        c = wmma4(a, Bp[k2 * bC + colS], c);
      }
    }
    const float NR = 0.05590170f;                 // 1/sqrt(320)
    float* out = res + row0 * 320 + outoff;
    #pragma unroll
    for (int i = 0; i < 8; i++) {
      int orow = i + 8 * half;
      out[orow * 320 + lr] = c[i] * NR;
    }
  }
}

// ---------------------------------------------------------------------------
// K2: fused radial MLP (WMMA) + tensor product + atomic scatter per 16 edges.
// One wave per 16-edge tile; h1/h2 staged via LDS; w consumed in C layout.
// agg layout per node (512 f32): [0:128]=agg_s, [128+m*128 + 0:128]=agg_v_m.
// ---------------------------------------------------------------------------
__global__ __launch_bounds__(128) void k_edge(
    const float* __restrict__ y, const float* __restrict__ esh,
    const int* __restrict__ snd, const int* __restrict__ rcv,
    const float* __restrict__ emb,
    const float* __restrict__ pWm1, const float* __restrict__ pWm2, const float* __restrict__ pWm3,
    float* __restrict__ agg)
{
  __shared__ float hA[EW][16][64];
  __shared__ float hB[EW][16][64];
  __shared__ int   msr[EW][16][2];
  __shared__ float msh[EW][16][4];

  const int lane = threadIdx.x & 31;
  const int wave = threadIdx.x >> 5;
  const int lr = lane & 15;
  const int half = lane >> 4;
  const int tile = blockIdx.x * EW + wave;
  const int ebase = tile * 16;

  if (lane < 16) {
    int e = ebase + lane;
    msr[wave][lane][0] = snd[e];
    msr[wave][lane][1] = rcv[e];
    #pragma unroll
    for (int q = 0; q < 4; q++) msh[wave][lane][q] = esh[e * 4 + q];
  }

  // stage 1: h1 = swish(emb @ Wm1 / sqrt(8)),  K = 8
  const float SC1 = 0.35355339f;
  const v2f* emb2 = (const v2f*)(emb + (ebase + lr) * 8);
  const v2f* W1p = (const v2f*)pWm1;
  for (int j = 0; j < 4; j++) {
    v8f c = vzero();
    #pragma unroll
    for (int t = 0; t < 2; t++) {
      int k2 = half + 2 * t;
      c = wmma4(emb2[k2], W1p[k2 * 64 + j * 16 + lr], c);
    }
    #pragma unroll
    for (int i = 0; i < 8; i++) {
      int r = i + 8 * half;
      hA[wave][r][j * 16 + lr] = swishf(c[i] * SC1);
    }
  }
  __syncthreads();

  // stage 2: h2 = swish(h1 @ Wm2 / 8),  K = 64
  const v2f* hA2 = (const v2f*)&hA[wave][lr][0];
  const v2f* W2p = (const v2f*)pWm2;
  for (int j = 0; j < 4; j++) {
    v8f c = vzero();
    #pragma unroll 4
    for (int t = 0; t < 16; t++) {
      int k2 = half + 2 * t;
      c = wmma4(hA2[k2], W2p[k2 * 64 + j * 16 + lr], c);
    }
    #pragma unroll
    for (int i = 0; i < 8; i++) {
      int r = i + 8 * half;
      hB[wave][r][j * 16 + lr] = swishf(c[i] * 0.125f);
    }
  }
  __syncthreads();

  // stage 3: w = h2 @ Wm3 / 8, then tensor product + scatter, tile by tile.
  const float INV_AVG = 0.0625f;       // 1/AVG_NEIGH
  const float ISQ3 = 0.57735027f;      // 1/sqrt(3)
  const v2f* hB2 = (const v2f*)&hB[wave][lr][0];
  const v2f* W3p = (const v2f*)pWm3;
  for (int j = 0; j < 16; j++) {
    v8f c = vzero();
    #pragma unroll 4
    for (int t = 0; t < 16; t++) {
      int k2 = half + 2 * t;
      c = wmma4(hB2[k2], W3p[k2 * 256 + j * 16 + lr], c);
    }
    const int u = (j & 3) * 16 + lr;   // feature within 0..63
    #pragma unroll
    for (int i = 0; i < 8; i++) {
      int r = i + 8 * half;            // edge row in tile
      float w = c[i] * 0.125f;
      int s  = msr[wave][r][0];
      int rv = msr[wave][r][1];
      float s0 = msh[wave][r][0];
      const float* ys = y + s * 256;
      float* ag = agg + rv * 512;
      if (j < 4) {                     // w1 * e0 * sh0 -> agg_s[0:64]
        atomicAdd(ag + u, w * ys[u] * s0 * INV_AVG);
      } else if (j < 8) {              // w2 * e0 * sh1_m -> agg_v_m[0:64]
        float e0v = ys[u] * w * INV_AVG;
        atomicAdd(ag + 128 + 0*128 + u, e0v * msh[wave][r][1]);
        atomicAdd(ag + 128 + 1*128 + u, e0v * msh[wave][r][2]);
        atomicAdd(ag + 128 + 2*128 + u, e0v * msh[wave][r][3]);
      } else if (j < 12) {             // w3 * e1_m * sh0 -> agg_v_m[64:128]
        float ws0 = w * s0 * INV_AVG;
        atomicAdd(ag + 128 + 0*128 + 64 + u, ws0 * ys[64 + 0*64 + u]);
        atomicAdd(ag + 128 + 1*128 + 64 + u, ws0 * ys[64 + 1*64 + u]);
        atomicAdd(ag + 128 + 2*128 + 64 + u, ws0 * ys[64 + 2*64 + u]);
      } else {                         // w4 * (e1.sh1)/sqrt3 -> agg_s[64:128]
        float dot = ys[64 + u]  * msh[wave][r][1]
                  + ys[128 + u] * msh[wave][r][2]
                  + ys[192 + u] * msh[wave][r][3];
        atomicAdd(ag + 64 + u, w * dot * ISQ3 * INV_AVG);
      }
    }
  }
}

// ---------------------------------------------------------------------------
// K3: z_s = agg_s @ W2s/sqrt(128) + res_s ; z_v_m = agg_v_m @ W2v/sqrt(128) + res_v_m
// grid (1250, 5) x 128 -> 20 wave-jobs per node tile.
// ---------------------------------------------------------------------------
__global__ __launch_bounds__(128) void k_post_mm(
    const float* __restrict__ agg, const float* __restrict__ res,
    const float* __restrict__ pW2s, const float* __restrict__ pW2v,
    float* __restrict__ z)
{
  const int lane = threadIdx.x & 31;
  const int wave = threadIdx.x >> 5;
  const int lr = lane & 15;
  const int half = lane >> 4;
  const int tile = blockIdx.x;
  const int job = blockIdx.y * 4 + wave;  // 0..19
  if (job >= 20) return;
  const int row0 = tile * 16;
  const int row = row0 + lr;
  const float N2 = 0.08838835f;           // 1/sqrt(128)

  const v2f* A2; const v2f* Bp; int bC, outoff;
  if (job < 8) {
    int j = job;
    A2 = (const v2f*)(agg + row * 512);
    Bp = (const v2f*)pW2s; bC = 128;
    outoff = j * 16;
  } else {
    int t = job - 8; int m = t >> 2; int j = t & 3;
    A2 = (const v2f*)(agg + row * 512 + 128 + m * 128);
    Bp = (const v2f*)pW2v; bC = 64;
    outoff = 128 + m * 64 + (t & 3) * 16;
  }
  const int col = (outoff & 63) >= 0 ? ((job < 8) ? (job * 16 + lr) : ((job - 8 & 3) * 16 + lr)) : 0;
  // recompute col cleanly:
  int colc = (job < 8) ? (job * 16 + lr) : (((job - 8) & 3) * 16 + lr);

  v8f c = vzero();
  #pragma unroll 4
  for (int t = 0; t < 32; t++) {
    int k2 = half + 2 * t;
    c = wmma4(A2[k2], Bp[k2 * bC + colc], c);
  }
  #pragma unroll
  for (int i = 0; i < 8; i++) {
    int orow = row0 + i + 8 * half;
    z[orow * 320 + outoff + lr] = c[i] * N2 + res[orow * 320 + outoff + lr];
  }
}

// ---------------------------------------------------------------------------
// K4: gating + output pack.  out = [swish(z_s[:,:64]), swish(z_s[:,64:])*z_v]
// ---------------------------------------------------------------------------
__global__ __launch_bounds__(256) void k_gate(const float* __restrict__ z,
                                              float* __restrict__ out)
{
  int idx = blockIdx.x * blockDim.x + threadIdx.x;   // N*64
  if (idx >= N_NODES * 64) return;
  int n = idx >> 6, u = idx & 63;
  const float* zr = z + n * 320;
  float os = swishf(zr[u]);
  float g  = swishf(zr[64 + u]);
  float* o = out + n * 256;
  o[u] = os;
  o[64 + u*3 + 0] = g * zr[128 + 0*64 + u];
  o[64 + u*3 + 1] = g * zr[128 + 1*64 + u];
  o[64 + u*3 + 2] = g * zr[128 + 2*64 + u];
}

extern "C" void kernel_launch(void* const* d_in, const int* in_sizes, int n_in,
                              void* d_out, int out_size, void* d_ws, size_t ws_size,
                              hipStream_t stream) {
  (void)in_sizes; (void)n_in; (void)out_size; (void)ws_size;
  const float* nf    = (const float*)d_in[0];
  const float* attrs = (const float*)d_in[1];
  const float* esh   = (const float*)d_in[2];
  const int*   snd   = (const int*)d_in[3];
  const int*   rcv   = (const int*)d_in[4];
  const float* emb   = (const float*)d_in[5];
  const float* Wl0   = (const float*)d_in[6];
  const float* Wl1   = (const float*)d_in[7];
  const float* Wm1   = (const float*)d_in[8];
  const float* Wm2   = (const float*)d_in[9];
  const float* Wm3   = (const float*)d_in[10];
  const float* W2s   = (const float*)d_in[11];
  const float* W2v   = (const float*)d_in[12];
  const float* Wrs   = (const float*)d_in[13];
  const float* Wrv   = (const float*)d_in[14];
  float* out = (float*)d_out;

  float* ws  = (float*)d_ws;
  float* y   = ws;                                     // N*256
  float* res = ws + (size_t)N_NODES*256;               // N*320
  float* agg = ws + (size_t)N_NODES*(256+320);         // N*512
  float* z   = ws + (size_t)N_NODES*(256+320+512);     // N*320
  float* pw  = ws + (size_t)N_NODES*(256+320+512+320); // pair-packed weights

  float* pWl0 = pw;            // 64x64
  float* pWl1 = pWl0 + 4096;   // 64x64
  float* pWm1 = pWl1 + 4096;   // 8x64
  float* pWm2 = pWm1 + 512;    // 64x64
  float* pWm3 = pWm2 + 4096;   // 64x256
  float* pW2s = pWm3 + 16384;  // 128x128
  float* pW2v = pW2s + 16384;  // 128x64
  float* pWrs = pW2v + 8192;   // 64x640  (u, s*128+v)
  float* pWrv = pWrs + 40960;  // 64x320  (u, s*64+v)

  PrepArgs pa;
  pa.src[0]=Wl0; pa.dst[0]=pWl0; pa.K[0]=64;  pa.C[0]=64;
  pa.src[1]=Wl1; pa.dst[1]=pWl1; pa.K[1]=64;  pa.C[1]=64;
  pa.src[2]=Wm1; pa.dst[2]=pWm1; pa.K[2]=8;   pa.C[2]=64;
  pa.src[3]=Wm2; pa.dst[3]=pWm2; pa.K[3]=64;  pa.C[3]=64;
  pa.src[4]=Wm3; pa.dst[4]=pWm3; pa.K[4]=64;  pa.C[4]=256;
  pa.src[5]=W2s; pa.dst[5]=pW2s; pa.K[5]=128; pa.C[5]=128;
  pa.src[6]=W2v; pa.dst[6]=pW2v; pa.K[6]=128; pa.C[6]=64;
  pa.src[7]=Wrs; pa.dst[7]=pWrs; pa.K[7]=64;  pa.C[7]=640;
  pa.src[8]=Wrv; pa.dst[8]=pWrv; pa.K[8]=64;  pa.C[8]=320;

  hipMemsetAsync(agg, 0, (size_t)N_NODES*512*sizeof(float), stream);
  k_prep<<<64, 256, 0, stream>>>(pa);
  k_node_pre<<<dim3(N_NODES/16, 9), 128, 0, stream>>>(nf, attrs, pWl0, pWl1, pWrs, pWrv, y, res);
  k_edge<<<dim3(N_EDGES/16/EW), 128, 0, stream>>>(y, esh, snd, rcv, emb, pWm1, pWm2, pWm3, agg);
  k_post_mm<<<dim3(N_NODES/16, 5), 128, 0, stream>>>(agg, res, pW2s, pW2v, z);
  k_gate<<<(N_NODES*64 + 255)/256, 256, 0, stream>>>(z, out);
}